// reshape_78271484002964
// MI455X (gfx1250) — compile-verified
//
#include <hip/hip_runtime.h>
#include <hip/hip_bf16.h>

// Pure layout permute: [16, 64, 256, 256] f32 -> [16, 2, 16, 16, 32, 16, 16].
// 512 MB total HBM traffic; ~22 us floor at 23.3 TB/s. Strategy: stage
// (ci=2, hj=16, k=16, wk=16) = 32 KB tiles through LDS so both global read
// and global write are 512B-contiguous per wave32 (full 128B HBM transactions),
// using CDNA5 async global->LDS loads, with an XOR bank swizzle so both LDS
// phases are conflict-free (each wave32 b128 op hits all 16 four-bank groups
// exactly twice = the 2-cycle minimum).

#if __has_builtin(__builtin_amdgcn_global_load_async_to_lds_b128)
#define HAVE_ASYNC_LDS 1
#endif

typedef int v4i __attribute__((ext_vector_type(4)));
#ifdef HAVE_ASYNC_LDS
typedef __attribute__((address_space(1))) v4i global_v4i;
typedef __attribute__((address_space(3))) v4i lds_v4i;
#endif

__global__ __launch_bounds__(256) void permute_split_kernel(
    const float* __restrict__ x, float* __restrict__ out) {
  __shared__ __align__(16) float lds[8192];  // 32 KB tile: 2048 float4 slots

  const int t = threadIdx.x;
  const unsigned blk = blockIdx.x;      // 8192 blocks = b(16) * i(2) * j(16) * cc(16)
  const int cc = blk & 15;              // channel chunk: covers ci0 = 2*cc .. 2*cc+1
  const int j  = (blk >> 4) & 15;
  const int i  = (blk >> 8) & 1;
  const int b  = (int)(blk >> 9);

  // flat float offsets (input stride: C->65536, H->256, W->1)
  const size_t in_base  = (size_t)b * 4194304u + (size_t)i * 2097152u
                        + (size_t)(cc * 2) * 65536u + (size_t)j * 4096u;
  // output strides: b->4194304, i->2097152, j->131072, k->8192, ci->256, hj->16, wk->1
  const size_t out_base = (size_t)b * 4194304u + (size_t)i * 2097152u
                        + (size_t)j * 131072u + (size_t)(cc * 2) * 256u;

  // ---- Phase 1: global -> LDS (rows contiguous in input) ----
  // 2048 float4s / 256 threads = 8 iterations. Row = (ci,hj): 256 contiguous
  // floats of input (all k,wk). LDS slot swizzled so phase 2 is conflict-free.
#pragma unroll
  for (int n = 0; n < 8; ++n) {
    const int f   = n * 256 + t;     // float4 id within tile, input-major
    const int row = f >> 6;          // 0..31  (ci*16 + hj)
    const int col = f & 63;          // 0..63  (k*4 + wk4)
    const int ci  = row >> 4;
    const int hj  = row & 15;
    const int k   = col >> 2;
    const int w4  = col & 3;
    const int p   = ci * 1024 + hj * 64 + (k >> 2) * 16
                  + (((k ^ hj) & 3) << 2) + w4;          // swizzled float4 slot
    const float* g = x + in_base + (size_t)ci * 65536u + (size_t)hj * 256u
                   + (size_t)(col * 4);
    float* l = &lds[4 * p];
#ifdef HAVE_ASYNC_LDS
    __builtin_amdgcn_global_load_async_to_lds_b128(
        (global_v4i*)g, (lds_v4i*)l, 0, 0);
#else
    *(float4*)l = *(const float4*)g;
#endif
  }

#ifdef HAVE_ASYNC_LDS
#if __has_builtin(__builtin_amdgcn_s_wait_asynccnt)
  __builtin_amdgcn_s_wait_asynccnt(0);
#else
  asm volatile("s_wait_asynccnt 0" ::: "memory");
#endif
#endif
  __syncthreads();

  // ---- Phase 2: LDS -> global (chunks contiguous in output) ----
  // Iteration n writes the 512-float contiguous output chunks for
  // k = 2n + (t>>7). Wave32 stores 512B contiguous.
  const int ci2 = (t >> 6) & 1;
  const int hj2 = (t >> 2) & 15;
  const int w42 = t & 3;
#pragma unroll
  for (int n = 0; n < 8; ++n) {
    const int k = n * 2 + (t >> 7);
    const int p = ci2 * 1024 + hj2 * 64 + (k >> 2) * 16
                + (((k ^ hj2) & 3) << 2) + w42;
    const float4 v = *(const float4*)&lds[4 * p];
    const size_t o = out_base + (size_t)k * 8192u + (size_t)ci2 * 256u
                   + (size_t)hj2 * 16u + (size_t)(w42 * 4);
    *(float4*)(out + o) = v;
  }
}

extern "C" void kernel_launch(void* const* d_in, const int* in_sizes, int n_in,
                              void* d_out, int out_size, void* d_ws, size_t ws_size,
                              hipStream_t stream) {
  (void)in_sizes; (void)n_in; (void)d_ws; (void)ws_size; (void)out_size;
  const float* x = (const float*)d_in[0];
  float* out = (float*)d_out;
  // 16(b) * 2(i) * 16(j) * 16(cc) = 8192 workgroups, 256 threads (8 wave32)
  permute_split_kernel<<<dim3(8192), dim3(256), 0, stream>>>(x, out);
}